// pointSIFT_res_module_87969520157196
// MI455X (gfx1250) — compile-verified
//
#include <hip/hip_runtime.h>
#include <math.h>

typedef float v2f __attribute__((ext_vector_type(2)));
typedef float v8f __attribute__((ext_vector_type(8)));

#define NB 8
#define NP 4096
#define NG (NB * NP)          // 32768 global points
#define OC 64

// ---------------------------------------------------------------------------
// Kernel 1: brute-force octant nearest-neighbor select (dominant cost, VALU)
// One thread per query point; candidates tiled through LDS.
// ---------------------------------------------------------------------------
__global__ __launch_bounds__(256) void sift_kernel(const float* __restrict__ xyz,
                                                   int* __restrict__ idx)
{
    __shared__ float sx[256], sy[256], sz[256];
    const int tid = threadIdx.x;
    const int blocksPerBatch = NP / 256;               // 16
    const int b = blockIdx.x / blocksPerBatch;
    const int i = (blockIdx.x % blocksPerBatch) * 256 + tid;
    const float* xb = xyz + (size_t)b * NP * 3;
    const float px = xb[i * 3 + 0];
    const float py = xb[i * 3 + 1];
    const float pz = xb[i * 3 + 2];

    float bd[8];
    int   bj[8];
#pragma unroll
    for (int k = 0; k < 8; ++k) { bd[k] = 3.402823466e38f; bj[k] = i; }

    for (int t = 0; t < NP; t += 256) {
        __syncthreads();
        sx[tid] = xb[(t + tid) * 3 + 0];
        sy[tid] = xb[(t + tid) * 3 + 1];
        sz[tid] = xb[(t + tid) * 3 + 2];
        __syncthreads();
#pragma unroll 4
        for (int jj = 0; jj < 256; ++jj) {
            const float dx = sx[jj] - px;
            const float dy = sy[jj] - py;
            const float dz = sz[jj] - pz;
            const float d2 = dx * dx + dy * dy + dz * dz;
            const int o = ((dx > 0.f) ? 4 : 0) + ((dy > 0.f) ? 2 : 0) + ((dz > 0.f) ? 1 : 0);
            const bool within = d2 < 0.25f;             // RADIUS^2
            const int j = t + jj;
#pragma unroll
            for (int k = 0; k < 8; ++k) {
                const bool take = within && (o == k) && (d2 < bd[k]);
                bd[k] = take ? d2 : bd[k];
                bj[k] = take ? j : bj[k];
            }
        }
    }
    int* op = idx + ((size_t)b * NP + i) * 8;
#pragma unroll
    for (int k = 0; k < 8; ++k) op[k] = bj[k];
}

// ---------------------------------------------------------------------------
// Kernel 2: gather-fused GEMM for layers 0 / 3 (K = 134, padded to 136).
//   y[row=(b,n,kp)][o] = sum_{c'} A[row][c'] * W[c'][o], BN + ReLU
//   A row = [xyz[j0]-ctr (3) | feat[j0] (64) | xyz[j1]-ctr (3) | feat[j1] (64)]
// The K-loop is segmented so 30 of 34 WMMA steps are branch-free contiguous
// feature loads; the 4 boundary steps use lane selects (no EXEC divergence).
// WMMA f32 16x16x4 lane mapping: A/B element K = v + 2*(lane>>4).
// ---------------------------------------------------------------------------
__global__ __launch_bounds__(256) void gemm_gather(const float* __restrict__ xyz,
                                                   const float* __restrict__ feat,   // (NG, 64)
                                                   const int*   __restrict__ idx,    // (NG, 8)
                                                   const float* __restrict__ W,      // (134, 64) row-major
                                                   const float* __restrict__ bng,
                                                   const float* __restrict__ bnb,
                                                   const float* __restrict__ bnm,
                                                   const float* __restrict__ bnv,
                                                   float* __restrict__ out)          // (NG*4, 64)
{
    const int lane = threadIdx.x & 31;
    const int hi   = lane >> 4;          // which half-wave (selects K pair)
    const int ln   = lane & 15;
    const int gw     = (blockIdx.x * blockDim.x + threadIdx.x) >> 5;
    const int nwaves = (gridDim.x * blockDim.x) >> 5;
    const int ntiles = (NG * 4) / 16;    // 8192

    // Per-lane B-fragment base: B[kb][col] with kb = kk + 2*hi + v
    //   -> Wb[kk*64 + v*64], Wb = W + col + 128*hi   (col = ct*16 + ln)
    const float* Wb = W + ln + 128 * hi;

    // BN constants depend only on the lane's column -> hoist out of tile loop
    float bns[4], bnmb[4], bnbt[4];
#pragma unroll
    for (int ct = 0; ct < 4; ++ct) {
        const int col = ct * 16 + ln;
        bns[ct]  = bng[col] / sqrtf(bnv[col] + 1e-3f);
        bnmb[ct] = bnm[col];
        bnbt[ct] = bnb[col];
    }

    for (int tile = gw; tile < ntiles; tile += nwaves) {
        const int arow = tile * 16 + ln;     // this lane's A row
        const int ng   = arow >> 2;          // global point index (b*NP+n)
        const int kp   = arow & 3;           // neighbor-pair index
        const int bb   = ng >> 12;           // batch
        const float cx = xyz[ng * 3 + 0];
        const float cy = xyz[ng * 3 + 1];
        const float cz = xyz[ng * 3 + 2];
        const int j0 = bb * NP + idx[ng * 8 + kp * 2 + 0];
        const int j1 = bb * NP + idx[ng * 8 + kp * 2 + 1];
        const float g0x = xyz[j0 * 3 + 0] - cx;
        const float g0y = xyz[j0 * 3 + 1] - cy;
        const float g0z = xyz[j0 * 3 + 2] - cz;
        const float g1x = xyz[j1 * 3 + 0] - cx;
        const float g1y = xyz[j1 * 3 + 1] - cy;
        const float g1z = xyz[j1 * 3 + 2] - cz;
        const float* f0 = feat + (size_t)j0 * 64;   // w=0 neighbor features
        const float* f1 = feat + (size_t)j1 * 64;   // w=1 neighbor features

        v8f acc[4];
#pragma unroll
        for (int ct = 0; ct < 4; ++ct)
#pragma unroll
            for (int r = 0; r < 8; ++r) acc[ct][r] = 0.f;

        // one WMMA K-step: A fragment `a`, weights at K offset kk
        auto step = [&](v2f a, int kk) {
#pragma unroll
            for (int ct = 0; ct < 4; ++ct) {
                v2f bf;
                bf[0] = Wb[kk * 64 + ct * 16];
                bf[1] = Wb[kk * 64 + ct * 16 + 64];
                acc[ct] = __builtin_amdgcn_wmma_f32_16x16x4_f32(
                    false, a, false, bf, (short)0, acc[ct], false, false);
            }
        };

        v2f a;
        // kk=0: c' = {0,1 | 2,3} -> {g0x,g0y | g0z, f0[0]}
        a[0] = hi ? g0z : g0x;
        a[1] = hi ? f0[0] : g0y;
        step(a, 0);
        // kk=4..60: c' = kk+2hi+v in [4,64) -> f0[c'-3] (contiguous, uniform)
#pragma unroll 5
        for (int kk = 4; kk < 64; kk += 4) {
            a[0] = f0[kk - 3 + 2 * hi];
            a[1] = f0[kk - 2 + 2 * hi];
            step(a, kk);
        }
        // kk=64: c' = {64,65 | 66,67} -> {f0[61],f0[62] | f0[63], g1x}
        a[0] = hi ? f0[63] : f0[61];
        a[1] = hi ? g1x : f0[62];
        step(a, 64);
        // kk=68: c' = {68,69 | 70,71} -> {g1y,g1z | f1[0],f1[1]}
        a[0] = hi ? f1[0] : g1y;
        a[1] = hi ? f1[1] : g1z;
        step(a, 68);
        // kk=72..128: c' in [72,132) -> f1[c'-70] (contiguous, uniform)
#pragma unroll 5
        for (int kk = 72; kk < 132; kk += 4) {
            a[0] = f1[kk - 70 + 2 * hi];
            a[1] = f1[kk - 69 + 2 * hi];
            step(a, kk);
        }
        // kk=132: c' = {132,133 | 134,135} -> {f1[62],f1[63] | pad 0, pad 0}
        a[0] = hi ? 0.f : f1[62];
        a[1] = hi ? 0.f : f1[63];
        {
#pragma unroll
            for (int ct = 0; ct < 4; ++ct) {
                v2f bf;
                bf[0] = hi ? 0.f : Wb[132 * 64 + ct * 16];
                bf[1] = hi ? 0.f : Wb[132 * 64 + ct * 16 + 64];
                acc[ct] = __builtin_amdgcn_wmma_f32_16x16x4_f32(
                    false, a, false, bf, (short)0, acc[ct], false, false);
            }
        }

        // BN + ReLU epilogue; lane column is ct*16+ln, rows r + 8*hi
#pragma unroll
        for (int ct = 0; ct < 4; ++ct) {
            const int col = ct * 16 + ln;
#pragma unroll
            for (int r = 0; r < 8; ++r) {
                const int row = tile * 16 + r + 8 * hi;
                float y = (acc[ct][r] - bnmb[ct]) * bns[ct] + bnbt[ct];
                y = fmaxf(y, 0.f);
                out[(size_t)row * 64 + col] = y;
            }
        }
    }
}

// ---------------------------------------------------------------------------
// Kernel 3: plain GEMM for layers 1/2/4/5: X (M,128) @ W (128,64), BN (+ReLU)
// ---------------------------------------------------------------------------
__global__ __launch_bounds__(256) void gemm_plain(const float* __restrict__ X,   // (M, 128)
                                                  const float* __restrict__ W,   // (128, 64)
                                                  const float* __restrict__ bng,
                                                  const float* __restrict__ bnb,
                                                  const float* __restrict__ bnm,
                                                  const float* __restrict__ bnv,
                                                  float* __restrict__ out,       // (M, 64)
                                                  int M, int relu)
{
    const int lane = threadIdx.x & 31;
    const int hi   = lane >> 4;
    const int ln   = lane & 15;
    const int gw     = (blockIdx.x * blockDim.x + threadIdx.x) >> 5;
    const int nwaves = (gridDim.x * blockDim.x) >> 5;
    const int ntiles = M / 16;

    const float* Wb = W + ln + 128 * hi;

    float bns[4], bnmb[4], bnbt[4];
#pragma unroll
    for (int ct = 0; ct < 4; ++ct) {
        const int col = ct * 16 + ln;
        bns[ct]  = bng[col] / sqrtf(bnv[col] + 1e-3f);
        bnmb[ct] = bnm[col];
        bnbt[ct] = bnb[col];
    }

    for (int tile = gw; tile < ntiles; tile += nwaves) {
        const int arow = tile * 16 + ln;
        const float* xr = X + (size_t)arow * 128 + 2 * hi;

        v8f acc[4];
#pragma unroll
        for (int ct = 0; ct < 4; ++ct)
#pragma unroll
            for (int r = 0; r < 8; ++r) acc[ct][r] = 0.f;

#pragma unroll 8
        for (int kk = 0; kk < 128; kk += 4) {
            const v2f a = *(const v2f*)(xr + kk);   // 8B aligned (kk, 2*hi even)
#pragma unroll
            for (int ct = 0; ct < 4; ++ct) {
                v2f bf;
                bf[0] = Wb[kk * 64 + ct * 16];
                bf[1] = Wb[kk * 64 + ct * 16 + 64];
                acc[ct] = __builtin_amdgcn_wmma_f32_16x16x4_f32(
                    false, a, false, bf, (short)0, acc[ct], false, false);
            }
        }
#pragma unroll
        for (int ct = 0; ct < 4; ++ct) {
            const int col = ct * 16 + ln;
#pragma unroll
            for (int r = 0; r < 8; ++r) {
                const int row = tile * 16 + r + 8 * hi;
                float y = (acc[ct][r] - bnmb[ct]) * bns[ct] + bnbt[ct];
                if (relu) y = fmaxf(y, 0.f);
                out[(size_t)row * 64 + col] = y;
            }
        }
    }
}

// ---------------------------------------------------------------------------
// Kernel 4: write tuple output (xyz, relu(y5+points), idx-as-float) flat
// ---------------------------------------------------------------------------
__global__ __launch_bounds__(256) void finalize_kernel(const float* __restrict__ xyz,
                                                       const float* __restrict__ points,
                                                       const float* __restrict__ y5,   // (NG,64)
                                                       const int*   __restrict__ idx,  // (NG,8)
                                                       float* __restrict__ out)
{
    const int t = blockIdx.x * blockDim.x + threadIdx.x;
    const int NXYZ = NG * 3;        // 98304
    const int NX   = NG * 64;       // 2097152
    const int NI   = NG * 8;        // 262144
    if (t < NXYZ) {
        out[t] = xyz[t];
    } else if (t < NXYZ + NX) {
        const int u = t - NXYZ;
        out[t] = fmaxf(y5[u] + points[u], 0.f);
    } else if (t < NXYZ + NX + NI) {
        const int u = t - NXYZ - NX;
        out[t] = (float)idx[u];
    }
}

// ---------------------------------------------------------------------------
extern "C" void kernel_launch(void* const* d_in, const int* in_sizes, int n_in,
                              void* d_out, int out_size, void* d_ws, size_t ws_size,
                              hipStream_t stream)
{
    const float* xyz    = (const float*)d_in[0];
    const float* points = (const float*)d_in[1];
    const float* W0     = (const float*)d_in[2];   // (2,67,64) == (134,64) flat
    const float* W1     = (const float*)d_in[3];   // (2,64,64) == (128,64) flat
    const float* W2     = (const float*)d_in[4];
    const float* W3     = (const float*)d_in[5];
    const float* W4     = (const float*)d_in[6];
    const float* W5     = (const float*)d_in[7];
    const float* bng    = (const float*)d_in[8];   // (6,64)
    const float* bnb    = (const float*)d_in[9];
    const float* bnm    = (const float*)d_in[10];
    const float* bnv    = (const float*)d_in[11];

    // Workspace layout (~60 MB total):
    char* ws = (char*)d_ws;
    int*   idx = (int*)ws;                                           //  1 MB
    float* y0  = (float*)(ws + (1u << 20));                          // (NG*4,64) 33.5 MB (also y3)
    float* y1  = (float*)(ws + (1u << 20) + 33554432u);              // (NG*2,64) 16.8 MB (also y4)
    float* y2  = (float*)(ws + (1u << 20) + 33554432u + 16777216u);  // (NG,64)    8.4 MB (also y5)

    // 1) octant nearest-neighbor selection
    sift_kernel<<<NB * (NP / 256), 256, 0, stream>>>(xyz, idx);

    // 2) stage-1 conv stack
    gemm_gather<<<512, 256, 0, stream>>>(xyz, points, idx, W0,
                                         bng + 0, bnb + 0, bnm + 0, bnv + 0, y0);
    gemm_plain<<<512, 256, 0, stream>>>(y0, W1, bng + 64,  bnb + 64,  bnm + 64,  bnv + 64,
                                        y1, NG * 2, 1);
    gemm_plain<<<512, 256, 0, stream>>>(y1, W2, bng + 128, bnb + 128, bnm + 128, bnv + 128,
                                        y2, NG, 1);

    // 3) stage-2 conv stack (re-gather y2 features; reuse buffers)
    gemm_gather<<<512, 256, 0, stream>>>(xyz, y2, idx, W3,
                                         bng + 192, bnb + 192, bnm + 192, bnv + 192, y0);
    gemm_plain<<<512, 256, 0, stream>>>(y0, W4, bng + 256, bnb + 256, bnm + 256, bnv + 256,
                                        y1, NG * 2, 1);
    gemm_plain<<<512, 256, 0, stream>>>(y1, W5, bng + 320, bnb + 320, bnm + 320, bnv + 320,
                                        y2, NG, 0);          // no ReLU on layer 5

    // 4) residual add + ReLU, emit (xyz, x, idx) flat
    const int total = NG * 3 + NG * 64 + NG * 8;              // 2,457,600
    finalize_kernel<<<(total + 255) / 256, 256, 0, stream>>>(xyz, points, y2, idx, (float*)d_out);
}